// PeakMoverLoss_11209864643166
// MI455X (gfx1250) — compile-verified
//
#include <hip/hip_runtime.h>
#include <math.h>

// ---- WMMA fragment types (gfx1250, wave32) ----
typedef __attribute__((ext_vector_type(2))) float v2f;   // A/B for 16x16x4 f32 (2 VGPRs)
typedef __attribute__((ext_vector_type(8))) float v8f;   // C/D 16x16 f32 (8 VGPRs)
// ---- TDM descriptor vector types ----
typedef __attribute__((ext_vector_type(4))) unsigned int v4u;
typedef __attribute__((ext_vector_type(8))) int v8i;
typedef __attribute__((ext_vector_type(4))) int v4i;

#define FDIM 300
#define ROWS 16          // rows per wave (= WMMA M)
#define OUT_STRIDE 308   // padded LDS stride: (8*308) mod 64 == 32 -> conflict-free halves
#define NTILES 19        // ceil(300/16) column tiles
#define KCHUNKS 6        // ceil((16+6)/4) K-chunks of 4 for the banded matmul
#define CHUNKS_PER_LANE 10  // ceil(300/32) positions per lane in stage 3

#if __has_builtin(__builtin_amdgcn_tensor_load_to_lds)
#define USE_TDM 1
#else
#define USE_TDM 0
#endif

__device__ __forceinline__ float coef_at(int i, float g0, float g1, float g2,
                                         float g3, float g4, float g5, float g6) {
    float r = 0.0f;
    r = (i == 0) ? g0 : r;
    r = (i == 1) ? g1 : r;
    r = (i == 2) ? g2 : r;
    r = (i == 3) ? g3 : r;
    r = (i == 4) ? g4 : r;
    r = (i == 5) ? g5 : r;
    r = (i == 6) ? g6 : r;
    return r;
}

__global__ __launch_bounds__(32)
void peak_mover_loss_kernel(const float* __restrict__ fr,
                            const float* __restrict__ freqs,
                            const float* __restrict__ gk,
                            float* __restrict__ out)
{
    __shared__ __align__(16) float ldsIn[ROWS * FDIM];        // 4800 f32: raw 16x300 tile
    __shared__ __align__(16) float ldsOut[ROWS * OUT_STRIDE]; // 4928 f32: blurred tile

    const int lane = threadIdx.x;            // 0..31 (wave32)
    const int r0   = blockIdx.x * ROWS;      // first row of this tile

    // ---- Stage 1: launch TDM DMA of the contiguous 16x300 f32 block (19.2 KB) ----
#if USE_TDM
    {
        // Tensor DMA: 1-D tile of 4800 f32 elements (block is fully contiguous).
        const unsigned long long gaddr =
            (unsigned long long)(const void*)(fr + (size_t)r0 * FDIM);
        const unsigned int lds_off = (unsigned int)(unsigned long long)(void*)ldsIn;
        const unsigned int NE = ROWS * FDIM;              // 4800 elements

        v4u g0;
        g0.x = 1u;                                        // count=1 (valid user descriptor)
        g0.y = lds_off;                                   // LDS byte address
        g0.z = (unsigned int)(gaddr & 0xFFFFFFFFull);     // global_addr[31:0]
        g0.w = (unsigned int)((gaddr >> 32) & 0x1FFFFFFull) | (2u << 30); // addr[56:32], type=2

        v8i g1;
        g1[0] = (int)(2u << 16);                          // data_size code 2 = 4 bytes
        g1[1] = (int)((NE & 0xFFFFu) << 16);              // tensor_dim0[15:0]  @ bits[79:48]
        g1[2] = (int)((NE >> 16) | (1u << 16));           // tensor_dim0[31:16], tensor_dim1=1
        g1[3] = (int)(NE << 16);                          // tensor_dim1 hi=0, tile_dim0=4800
        g1[4] = 1;                                        // tile_dim1=1, tile_dim2=0
        g1[5] = (int)NE;                                  // tensor_dim0_stride lo (unused for 1D)
        g1[6] = 0;
        g1[7] = 0;

        v4i gz4 = {0, 0, 0, 0};
        v8i gz8 = {0, 0, 0, 0, 0, 0, 0, 0};
        // 6-arg toolchain form (clang-23 / therock-10.0): (g0, g1, g2, g3, g4, cpol)
        __builtin_amdgcn_tensor_load_to_lds(g0, g1, gz4, gz4, gz8, 0);
        // NOTE: s_wait_tensorcnt deferred until just before first LDS consumer,
        // so the setup work below overlaps with the DMA.
    }
#else
    {
        const float4* __restrict__ src4 = reinterpret_cast<const float4*>(fr + (size_t)r0 * FDIM);
        float4* dst4 = reinterpret_cast<float4*>(ldsIn);
        #pragma unroll 4
        for (int i = lane; i < (ROWS * FDIM) / 4; i += 32) dst4[i] = src4[i];
    }
#endif

    // ---- Setup work overlapped with the tensor DMA ----
    // Gaussian taps (uniform broadcast loads, hot in cache)
    const float g0 = gk[0], g1 = gk[1], g2 = gk[2], g3 = gk[3],
                g4 = gk[4], g5 = gk[5], g6 = gk[6];

    // freqs chunk per lane, reused by all 16 rows
    float fq[CHUNKS_PER_LANE];
    #pragma unroll
    for (int k = 0; k < CHUNKS_PER_LANE; ++k) {
        const int c = lane + 32 * k;
        fq[k] = (c < FDIM) ? freqs[c] : 0.0f;
    }

    // ---- Stage 2 constants: banded-B fragments for V_WMMA_F32_16X16X4_F32 ----
    // A (16x4 f32): lane L -> row M = L&15; VGPR0 -> K = 2*(L>>4), VGPR1 -> K+1.
    // B (4x16 f32): lane L -> col N = L&15; VGPR0 -> K = 2*(L>>4), VGPR1 -> K+1.
    const int m  = lane & 15;
    const int kh = (lane >> 4) * 2;

    // B[k][n] couples input col (j0-3+kc*4+k) to output col (j0+n): tap idx = kc*4+k-n.
    // Pin fragments in VGPRs with an opaque asm barrier so the select chains are NOT
    // rematerialized inside the tile loop.
    v2f bfrag[KCHUNKS];
    #pragma unroll
    for (int kc = 0; kc < KCHUNKS; ++kc) {
        const int i0 = kc * 4 + kh - m;
        float bx = coef_at(i0,     g0, g1, g2, g3, g4, g5, g6);
        float by = coef_at(i0 + 1, g0, g1, g2, g3, g4, g5, g6);
        asm volatile("" : "+v"(bx), "+v"(by));   // keep resident; block remat
        bfrag[kc].x = bx;
        bfrag[kc].y = by;
    }

#if USE_TDM
    __builtin_amdgcn_s_wait_tensorcnt(0);        // DMA complete before LDS reads
#endif
    __syncthreads();

    // ---- Stage 2: blur = banded matmul via V_WMMA_F32_16X16X4_F32 ----
    for (int t = 0; t < NTILES; ++t) {
        const int j0 = t * 16;
        v8f acc = {};
        #pragma unroll
        for (int kc = 0; kc < KCHUNKS; ++kc) {
            const int col = j0 - 3 + kc * 4 + kh;
            v2f a;
            a.x = ((unsigned)col       < (unsigned)FDIM) ? ldsIn[m * FDIM + col]     : 0.0f;
            a.y = ((unsigned)(col + 1) < (unsigned)FDIM) ? ldsIn[m * FDIM + col + 1] : 0.0f;
            acc = __builtin_amdgcn_wmma_f32_16x16x4_f32(
                false, a, false, bfrag[kc], (short)0, acc, false, false);
        }
        // D: lanes 0-15 VGPR v -> (M=v, N=lane); lanes 16-31 -> (M=v+8, N=lane-16)
        const int orow = (lane >> 4) * 8;
        const int ocol = j0 + m;
        if (ocol < FDIM) {
            #pragma unroll
            for (int v = 0; v < 8; ++v)
                ldsOut[(orow + v) * OUT_STRIDE + ocol] = acc[v];
        }
    }

    __syncthreads();

    // ---- Stage 3: per-row first-two-peaks + masked softargmax ----
    for (int r = 0; r < ROWS; ++r) {
        const float* f = ldsOut + r * OUT_STRIDE;

        // Pass 1: cache row chunk in registers; peaks + FULL-ROW max in one pass.
        // (softmax is shift-invariant, so the row max works in place of the masked max)
        float fv[CHUNKS_PER_LANE];
        float mx = -INFINITY;
        int p0 = FDIM, p1 = FDIM;
        #pragma unroll
        for (int k = 0; k < CHUNKS_PER_LANE; ++k) {
            const int c = lane + 32 * k;
            const bool in = (c < FDIM);
            const float fc = in ? f[c] : -INFINITY;
            fv[k] = fc;
            mx = fmaxf(mx, fc);
            if (in) {
                const float fm1 = f[(c > 0) ? (c - 1) : 0];
                const float fp1 = f[(c < FDIM - 1) ? (c + 1) : (FDIM - 1)];
                const bool left_ok  = (c == 0)        || (fc > fm1);
                const bool right_ok = (c == FDIM - 1) || (fc > fp1);
                if (left_ok && right_ok) {
                    if (p0 == FDIM) p0 = c; else if (p1 == FDIM) p1 = c;
                }
            }
        }
        // Fused tree: two-smallest merge (pairs sorted per lane) + max, 5 steps.
        #pragma unroll
        for (int off = 16; off > 0; off >>= 1) {
            const int   q0 = __shfl_down(p0, off, 32);
            const int   q1 = __shfl_down(p1, off, 32);
            const float qm = __shfl_down(mx, off, 32);
            const int   n1 = min(max(p0, q0), min(p1, q1));
            p0 = min(p0, q0);
            p1 = n1;
            mx = fmaxf(mx, qm);
        }
        p0 = __shfl(p0, 0, 32);
        p1 = __shfl(p1, 0, 32);
        mx = __shfl(mx, 0, 32);
        const int end = (p1 < FDIM) ? ((p0 + p1) >> 1) : (FDIM - 1);  // end >= 1 always

        // Pass 2: register-only exp-sum and freq-weighted sum over c < end.
        float s = 0.0f, w = 0.0f;
        #pragma unroll
        for (int k = 0; k < CHUNKS_PER_LANE; ++k) {
            const int c = lane + 32 * k;
            const float e = (c < end) ? __expf(fv[k] - mx) : 0.0f;
            s += e;
            w += e * fq[k];
        }
        #pragma unroll
        for (int off = 16; off > 0; off >>= 1) {
            s += __shfl_down(s, off, 32);
            w += __shfl_down(w, off, 32);
        }
        if (lane == 0) out[r0 + r] = -(w / s);
    }
}

extern "C" void kernel_launch(void* const* d_in, const int* in_sizes, int n_in,
                              void* d_out, int out_size, void* d_ws, size_t ws_size,
                              hipStream_t stream) {
    const float* fr    = (const float*)d_in[0];  // (B, 300) f32
    const float* freqs = (const float*)d_in[1];  // (300,)   f32
    const float* gk    = (const float*)d_in[2];  // (7,)     f32
    float* out = (float*)d_out;                  // (B,)     f32

    const int B = in_sizes[0] / FDIM;            // 65536
    dim3 grid(B / ROWS);                         // 4096 one-wave workgroups
    dim3 block(32);
    peak_mover_loss_kernel<<<grid, block, 0, stream>>>(fr, freqs, gk, out);
}